// Net_18915035971604
// MI455X (gfx1250) — compile-verified
//
#include <hip/hip_runtime.h>
#include <cstddef>

#define N_NODES 20000
#define N_EDGES 640000
#define NB 10
#define OUT_CH 6890
#define MCHUNK 64

typedef float v2f __attribute__((ext_vector_type(2)));
typedef float v8f __attribute__((ext_vector_type(8)));

__device__ __forceinline__ float selu_f(float x) {
    const float alpha = 1.6732632423543772f;
    const float sc    = 1.0507009873554805f;
    return x > 0.0f ? sc * x : sc * alpha * (expf(x) - 1.0f);
}

__global__ void k_zero_i32(int* __restrict__ p, int n) {
    int i = blockIdx.x * blockDim.x + threadIdx.x;
    if (i < n) p[i] = 0;
}

// Per-edge: 10-term basis from rel = (pos[src]-pos[dst])/R_FACTOR, count in-degree.
__global__ void k_edge_prep(const int* __restrict__ ei, const float* __restrict__ pos,
                            float* __restrict__ basis, int* __restrict__ counts) {
    int e = blockIdx.x * blockDim.x + threadIdx.x;
    if (e >= N_EDGES) return;
    int s = ei[e];
    int d = ei[N_EDGES + e];
    const float inv = (float)(1.0 / 2.4031); // R_FACTOR = 1.1599 - (-1.2432)
    float dx = (pos[3*s+0] - pos[3*d+0]) * inv;
    float dy = (pos[3*s+1] - pos[3*d+1]) * inv;
    float dz = (pos[3*s+2] - pos[3*d+2]) * inv;
    float* bs = basis + (size_t)e * NB;
    bs[0] = 1.0f; bs[1] = dx;    bs[2] = dy;    bs[3] = dz;    bs[4] = dx*dx;
    bs[5] = dx*dy; bs[6] = dx*dz; bs[7] = dy*dy; bs[8] = dy*dz; bs[9] = dz*dz;
    atomicAdd(&counts[d], 1);
}

// Single-block inclusive scan: counts -> CSR row_start; deg_inv = 1/max(deg,1).
__global__ __launch_bounds__(1024)
void k_scan(const int* __restrict__ counts, int* __restrict__ row_start,
            float* __restrict__ deg_inv) {
    __shared__ int sdata[1024];
    __shared__ int carry;
    int tid = threadIdx.x;
    if (tid == 0) { carry = 0; row_start[0] = 0; }
    __syncthreads();
    for (int base = 0; base < N_NODES; base += 1024) {
        int i = base + tid;
        int v = (i < N_NODES) ? counts[i] : 0;
        if (i < N_NODES) deg_inv[i] = 1.0f / fmaxf((float)v, 1.0f);
        sdata[tid] = v;
        __syncthreads();
        for (int off = 1; off < 1024; off <<= 1) {
            int t = (tid >= off) ? sdata[tid - off] : 0;
            __syncthreads();
            sdata[tid] += t;
            __syncthreads();
        }
        if (i < N_NODES) row_start[i + 1] = carry + sdata[tid];
        __syncthreads();
        if (tid == 0) carry += sdata[1023];
        __syncthreads();
    }
}

// Bucket edges by dst.
__global__ void k_edge_fill(const int* __restrict__ ei, const int* __restrict__ row_start,
                            int* __restrict__ cursor, int* __restrict__ edge_perm) {
    int e = blockIdx.x * blockDim.x + threadIdx.x;
    if (e >= N_EDGES) return;
    int d = ei[N_EDGES + e];
    int p = atomicAdd(&cursor[d], 1);
    edge_perm[row_start[d] + p] = e;
}

// One block per node. Edge ids / src / basis staged through LDS in 64-edge chunks;
// thread t = channel c accumulates 10 moments in registers.
// Writes M[n, b*cin + c] with row stride Kp (pad zeroed).
__global__ __launch_bounds__(128)
void k_moment(const float* __restrict__ h, const float* __restrict__ basis,
              const int* __restrict__ src, const int* __restrict__ edge_perm,
              const int* __restrict__ row_start, float* __restrict__ M,
              int cin, int Kp) {
    __shared__ int   s_eid[MCHUNK];
    __shared__ int   s_src[MCHUNK];
    __shared__ float s_bas[MCHUNK * NB];
    const int n   = blockIdx.x;
    const int tid = threadIdx.x;
    const int c   = tid;
    const int j0 = row_start[n], j1 = row_start[n + 1];
    float acc[NB];
#pragma unroll
    for (int b = 0; b < NB; ++b) acc[b] = 0.0f;

    for (int jc = j0; jc < j1; jc += MCHUNK) {
        const int cnt = min(MCHUNK, j1 - jc);
        __syncthreads();
        if (tid < cnt) {
            int e = edge_perm[jc + tid];
            s_eid[tid] = e;
            s_src[tid] = src[e];
        }
        __syncthreads();
        for (int t = tid; t < cnt * NB; t += 128) {
            int ii = t / NB;
            int b  = t - ii * NB;
            s_bas[t] = basis[(size_t)s_eid[ii] * NB + b];
        }
        __syncthreads();
        if (c < cin) {
            for (int ii = 0; ii < cnt; ++ii) {
                float xv = h[(size_t)s_src[ii] * cin + c];
                const float* bs = &s_bas[ii * NB];
#pragma unroll
                for (int b = 0; b < NB; ++b) acc[b] = fmaf(bs[b], xv, acc[b]);
            }
        }
    }
    float* Mrow = M + (size_t)n * Kp;
    if (c < cin) {
#pragma unroll
        for (int b = 0; b < NB; ++b) Mrow[b * cin + c] = acc[b];
    }
    const int K = NB * cin;
    if (tid < Kp - K) Mrow[K + tid] = 0.0f;   // zero K-pad (layer 0 only)
}

// Repack B(KxN) row-major -> Bp: pair-interleaved, zero-padded to Kp x Np.
// Bp[((k>>1)*Np + n)*2 + (k&1)] so a WMMA B-fragment is one contiguous b64 load.
__global__ void k_pack_b(const float* __restrict__ B, float* __restrict__ Bp,
                         int K, int N, int Kp, int Np) {
    int i = blockIdx.x * blockDim.x + threadIdx.x;
    if (i >= Kp * Np) return;
    int k = i / Np;
    int n = i - k * Np;
    float v = (k < K && n < N) ? B[(size_t)k * N + n] : 0.0f;
    Bp[((size_t)(k >> 1) * Np + n) * 2 + (k & 1)] = v;
}

// fp32 WMMA GEMM: C(Mr x N) = epilogue(A(Mr x Kp, zero-padded) @ B(Kp x Np packed)).
// Wave computes 32x64 via 8 V_WMMA_F32_16X16X4_F32 accumulators (2 A frags x 4 B frags).
// Software-pipelined: next K-step fragments load while current WMMAs execute.
__global__ __launch_bounds__(256)
void k_wmma_gemm(const float* __restrict__ A, const float* __restrict__ Bp,
                 float* __restrict__ C, int Mr, int Kp, int N, int Np,
                 const float* __restrict__ bias, const float* __restrict__ scale,
                 int do_selu) {
    const int wave = threadIdx.x >> 5;
    const int lane = threadIdx.x & 31;
    const int half = lane >> 4;   // K-pair select: {0,1} vs {2,3}
    const int r    = lane & 15;   // M index (A frag) / N index (B frag)
    const int row0 = (blockIdx.y * 8 + wave) * 32;
    if (row0 >= Mr) return;       // wave-uniform; Mr % 32 == 0 so no row guards below
    const int col0 = blockIdx.x * 64;

    v8f acc[8] = {};
    const float* a_ptr0 = A + (size_t)(row0 + r) * Kp + 2 * half;
    const float* a_ptr1 = a_ptr0 + (size_t)16 * Kp;
    const float* b_ptr0 = Bp + ((size_t)half * Np + (col0 +  0 + r)) * 2;
    const float* b_ptr1 = Bp + ((size_t)half * Np + (col0 + 16 + r)) * 2;
    const float* b_ptr2 = Bp + ((size_t)half * Np + (col0 + 32 + r)) * 2;
    const float* b_ptr3 = Bp + ((size_t)half * Np + (col0 + 48 + r)) * 2;
    const size_t bstep = (size_t)4 * Np;  // floats per 4-wide K step

    const int steps = Kp >> 2;            // >= 8 for all our shapes
    v2f a0 = *(const v2f*)a_ptr0;
    v2f a1 = *(const v2f*)a_ptr1;
    v2f b0 = *(const v2f*)b_ptr0;
    v2f b1 = *(const v2f*)b_ptr1;
    v2f b2 = *(const v2f*)b_ptr2;
    v2f b3 = *(const v2f*)b_ptr3;

    for (int s = 0; s < steps - 1; ++s) {
        a_ptr0 += 4; a_ptr1 += 4;
        b_ptr0 += bstep; b_ptr1 += bstep; b_ptr2 += bstep; b_ptr3 += bstep;
        v2f na0 = *(const v2f*)a_ptr0;
        v2f na1 = *(const v2f*)a_ptr1;
        v2f nb0 = *(const v2f*)b_ptr0;
        v2f nb1 = *(const v2f*)b_ptr1;
        v2f nb2 = *(const v2f*)b_ptr2;
        v2f nb3 = *(const v2f*)b_ptr3;
        acc[0] = __builtin_amdgcn_wmma_f32_16x16x4_f32(false, a0, false, b0, (short)0, acc[0], false, false);
        acc[4] = __builtin_amdgcn_wmma_f32_16x16x4_f32(false, a1, false, b0, (short)0, acc[4], false, false);
        acc[1] = __builtin_amdgcn_wmma_f32_16x16x4_f32(false, a0, false, b1, (short)0, acc[1], false, false);
        acc[5] = __builtin_amdgcn_wmma_f32_16x16x4_f32(false, a1, false, b1, (short)0, acc[5], false, false);
        acc[2] = __builtin_amdgcn_wmma_f32_16x16x4_f32(false, a0, false, b2, (short)0, acc[2], false, false);
        acc[6] = __builtin_amdgcn_wmma_f32_16x16x4_f32(false, a1, false, b2, (short)0, acc[6], false, false);
        acc[3] = __builtin_amdgcn_wmma_f32_16x16x4_f32(false, a0, false, b3, (short)0, acc[3], false, false);
        acc[7] = __builtin_amdgcn_wmma_f32_16x16x4_f32(false, a1, false, b3, (short)0, acc[7], false, false);
        a0 = na0; a1 = na1; b0 = nb0; b1 = nb1; b2 = nb2; b3 = nb3;
    }
    acc[0] = __builtin_amdgcn_wmma_f32_16x16x4_f32(false, a0, false, b0, (short)0, acc[0], false, false);
    acc[4] = __builtin_amdgcn_wmma_f32_16x16x4_f32(false, a1, false, b0, (short)0, acc[4], false, false);
    acc[1] = __builtin_amdgcn_wmma_f32_16x16x4_f32(false, a0, false, b1, (short)0, acc[1], false, false);
    acc[5] = __builtin_amdgcn_wmma_f32_16x16x4_f32(false, a1, false, b1, (short)0, acc[5], false, false);
    acc[2] = __builtin_amdgcn_wmma_f32_16x16x4_f32(false, a0, false, b2, (short)0, acc[2], false, false);
    acc[6] = __builtin_amdgcn_wmma_f32_16x16x4_f32(false, a1, false, b2, (short)0, acc[6], false, false);
    acc[3] = __builtin_amdgcn_wmma_f32_16x16x4_f32(false, a0, false, b3, (short)0, acc[3], false, false);
    acc[7] = __builtin_amdgcn_wmma_f32_16x16x4_f32(false, a1, false, b3, (short)0, acc[7], false, false);

#pragma unroll
    for (int t = 0; t < 4; ++t) {
        const int col = col0 + t * 16 + r;
        if (col < N) {
            const float bv = bias ? bias[col] : 0.0f;
#pragma unroll
            for (int rt = 0; rt < 2; ++rt) {
#pragma unroll
                for (int v = 0; v < 8; ++v) {
                    const int row = row0 + rt * 16 + v + 8 * half;
                    float val = acc[t + 4 * rt][v];
                    if (scale) val *= scale[row];
                    val += bv;
                    if (do_selu) val = selu_f(val);
                    C[(size_t)row * N + col] = val;
                }
            }
        }
    }
}

static inline int ceil4(int x)  { return (x + 3)  & ~3;  }
static inline int ceil64(int x) { return (x + 63) & ~63; }

extern "C" void kernel_launch(void* const* d_in, const int* in_sizes, int n_in,
                              void* d_out, int out_size, void* d_ws, size_t ws_size,
                              hipStream_t stream) {
    static const int CH[9] = {3, 16, 32, 64, 128, 128, 128, 128, 128};
    const float* x   = (const float*)d_in[0];
    const int*   ei  = (const int*)d_in[1];
    const float* pos = (const float*)d_in[2];
    const float* Wl[8]; const float* bl[8];
    const float *fc1W, *fc1b, *fc2W, *fc2b;
    if (n_in >= 23) {          // tuples flattened into separate inputs
        for (int i = 0; i < 8; ++i) { Wl[i] = (const float*)d_in[3 + i]; bl[i] = (const float*)d_in[11 + i]; }
        fc1W = (const float*)d_in[19]; fc1b = (const float*)d_in[20];
        fc2W = (const float*)d_in[21]; fc2b = (const float*)d_in[22];
    } else {                   // tuples passed as concatenated buffers
        const float* Wc = (const float*)d_in[3];
        const float* bc = (const float*)d_in[4];
        size_t wo = 0, bo = 0;
        for (int i = 0; i < 8; ++i) {
            Wl[i] = Wc + wo; wo += (size_t)NB * CH[i] * CH[i + 1];
            bl[i] = bc + bo; bo += CH[i + 1];
        }
        fc1W = (const float*)d_in[5]; fc1b = (const float*)d_in[6];
        fc2W = (const float*)d_in[7]; fc2b = (const float*)d_in[8];
    }

    // Workspace carve (every scratch byte read is written first each call).
    char* w = (char*)d_ws;
    auto carve = [&](size_t bytes) -> char* {
        char* p = w; w += (bytes + 255) & ~(size_t)255; return p;
    };
    float* basis    = (float*)carve((size_t)N_EDGES * NB * 4);        // 25.6 MB
    int*   counts   = (int*)  carve((size_t)N_NODES * 4);
    int*   cursor   = (int*)  carve((size_t)N_NODES * 4);
    int*   rowstart = (int*)  carve((size_t)(N_NODES + 1) * 4);
    int*   eperm    = (int*)  carve((size_t)N_EDGES * 4);             // 2.56 MB
    float* deginv   = (float*)carve((size_t)N_NODES * 4);
    float* Mbuf     = (float*)carve((size_t)N_NODES * NB * 128 * 4);  // 102.4 MB
    float* hA       = (float*)carve((size_t)N_NODES * 256 * 4);       // 20.5 MB
    float* hB       = (float*)carve((size_t)N_NODES * 256 * 4);       // 20.5 MB
    float* Bpack    = (float*)carve((size_t)256 * 6912 * 4);          // 7.1 MB (max Kp*Np)

    const int tb = 256;
    k_zero_i32<<<(N_NODES + tb - 1) / tb, tb, 0, stream>>>(counts, N_NODES);
    k_zero_i32<<<(N_NODES + tb - 1) / tb, tb, 0, stream>>>(cursor, N_NODES);
    k_edge_prep<<<(N_EDGES + tb - 1) / tb, tb, 0, stream>>>(ei, pos, basis, counts);
    k_scan<<<1, 1024, 0, stream>>>(counts, rowstart, deginv);
    k_edge_fill<<<(N_EDGES + tb - 1) / tb, tb, 0, stream>>>(ei, rowstart, cursor, eperm);

    const int gy = (N_NODES + 255) / 256;   // 8 waves x 32 rows per block

    auto run_gemm = [&](const float* A, int Kp, const float* B, int K, int N,
                        const float* bias, const float* scale, int selu, float* out) {
        const int Np = ceil64(N);
        k_pack_b<<<((Kp * Np) + tb - 1) / tb, tb, 0, stream>>>(B, Bpack, K, N, Kp, Np);
        dim3 g(Np / 64, gy);
        k_wmma_gemm<<<g, 256, 0, stream>>>(A, Bpack, out, N_NODES, Kp, N, Np, bias, scale, selu);
    };

    const float* hin = x;
    float* cur = hA;
    float* nxt = hB;
    for (int li = 0; li < 8; ++li) {
        const int cin = CH[li], cout = CH[li + 1];
        const int K  = NB * cin;
        const int Kp = ceil4(K);
        k_moment<<<N_NODES, 128, 0, stream>>>(hin, basis, ei, eperm, rowstart, Mbuf, cin, Kp);
        run_gemm(Mbuf, Kp, Wl[li], K, cout, bl[li], deginv, 1, cur);
        hin = cur;
        float* t = cur; cur = nxt; nxt = t;
    }
    // fc1: (20000x128)@(128x256) + b, SELU   (stride 128 already 4-aligned)
    run_gemm(hin, 128, fc1W, 128, 256, fc1b, nullptr, 1, cur);
    // fc2: (20000x256)@(256x6890) + b -> d_out
    run_gemm(cur, 256, fc2W, 256, OUT_CH, fc2b, nullptr, 0, (float*)d_out);
}